// MemoryClsEuclideanRW_45681272160457
// MI455X (gfx1250) — compile-verified
//
#include <hip/hip_runtime.h>
#include <hip/hip_bf16.h>

// ---------------------------------------------------------------------------
// MemoryClsEuclideanRW for MI455X (gfx1250, wave32, WMMA + async global->LDS)
//
// z:      (64, 512, 16, 16) f32      d_in[0]
// cls:    (64,) i32                  d_in[1]
// memory: (1, 512, 2000, 10) f32     d_in[2]
// out:    z_hat (64,512,16,16) | w_hat (64,256,2000) | w | log_w  (all f32)
// ---------------------------------------------------------------------------

#define NN     64
#define CC     512
#define HW     256
#define MM     2000
#define NCLS   10
#define CM     (CC * MM)          // 1,024,000 elements per class plane

typedef float v2f __attribute__((ext_vector_type(2)));
typedef float v8f __attribute__((ext_vector_type(8)));

#define AS1 __attribute__((address_space(1)))
#define AS3 __attribute__((address_space(3)))

// async global -> LDS dword copy (per-lane addresses), tracked by ASYNCcnt
__device__ __forceinline__ void async_g2l_b32(const float* g, float* l) {
    __builtin_amdgcn_global_load_async_to_lds_b32(
        (AS1 int*)(AS1 const void*)g, (AS3 int*)(AS3 void*)l, 0, 0);
}
__device__ __forceinline__ void async_wait0() {
    __builtin_amdgcn_s_wait_asynccnt(0);
}

// ---------------------------------------------------------------------------
// Kernel 1: gather memory[0,:,:,cls] into per-class [c][m] planes (coalesced)
// memT_cm[cl][c*MM + m] = memory[((c*MM)+m)*10 + cl]
// ---------------------------------------------------------------------------
__global__ void k_gather_cm(const float* __restrict__ mem, float* __restrict__ memT_cm) {
    int idx = blockIdx.x * blockDim.x + threadIdx.x;   // (c, m) flat
    if (idx >= CM) return;
    const float* src = mem + (size_t)idx * NCLS;       // 10 contiguous floats
#pragma unroll
    for (int cl = 0; cl < NCLS; ++cl)
        memT_cm[(size_t)cl * CM + idx] = src[cl];
}

// ---------------------------------------------------------------------------
// Kernel 2: tiled transpose per class: memT_mc[cl][m*CC + c] = memT_cm[cl][c*MM + m]
// ---------------------------------------------------------------------------
__global__ void k_transpose_mc(const float* __restrict__ cm, float* __restrict__ mc) {
    __shared__ float tile[32][33];
    int cl = blockIdx.z;
    int m0 = blockIdx.x * 32;
    int c0 = blockIdx.y * 32;
    const float* src = cm + (size_t)cl * CM;
    float*       dst = mc + (size_t)cl * CM;
    int x = threadIdx.x;                               // 0..31
    for (int i = threadIdx.y; i < 32; i += 8) {
        int m = m0 + x, c = c0 + i;
        if (m < MM) tile[i][x] = src[(size_t)c * MM + m];
    }
    __syncthreads();
    for (int i = threadIdx.y; i < 32; i += 8) {
        int m = m0 + i, c = c0 + x;
        if (m < MM) dst[(size_t)m * CC + c] = tile[x][i];
    }
}

// ---------------------------------------------------------------------------
// Kernel 3: m2[cl][m] = sum_c memT_cm[cl][c][m]^2   (coalesced over m)
// ---------------------------------------------------------------------------
__global__ void k_m2(const float* __restrict__ cm, float* __restrict__ m2) {
    int idx = blockIdx.x * blockDim.x + threadIdx.x;   // cl*MM + m
    if (idx >= NCLS * MM) return;
    int cl = idx / MM, m = idx % MM;
    const float* p = cm + (size_t)cl * CM + m;
    float acc = 0.f;
    for (int c = 0; c < CC; ++c) { float v = p[(size_t)c * MM]; acc += v * v; }
    m2[idx] = acc;
}

// ---------------------------------------------------------------------------
// Kernel 4: z2[n][p] = sum_c z[n,c,p]^2   (coalesced over p)
// ---------------------------------------------------------------------------
__global__ void k_z2(const float* __restrict__ z, float* __restrict__ z2) {
    int n = blockIdx.x, p = threadIdx.x;
    const float* q = z + ((size_t)n * CC) * HW + p;
    float acc = 0.f;
    for (int c = 0; c < CC; ++c) { float v = q[(size_t)c * HW]; acc += v * v; }
    z2[n * HW + p] = acc;
}

// ---------------------------------------------------------------------------
// Kernel 5: fused GEMM1 + softmax.
// Block = (sample n, 16-row p-tile).  256 threads = 8 waves.
// sim[p, m] = 2 * sum_c z[n,c,p]*memcls[c,m] - z2[p] - m2[m]
// A-panel staged via async global->LDS; each wave owns M-tiles {wave + 8*i},
// keeps its full sim strip in registers, does row max / exp-sum reductions
// (shfl within 16-lane halves, LDS across waves), writes w, w_hat, log_w.
// ---------------------------------------------------------------------------
__global__ void __launch_bounds__(256)
k_gemm1_softmax(const float* __restrict__ z, const int* __restrict__ cls,
                const float* __restrict__ memT_cm, const float* __restrict__ m2g,
                const float* __restrict__ z2g,
                float* __restrict__ out_what, float* __restrict__ out_w,
                float* __restrict__ out_logw) {
    __shared__ float As[CC * 16];       // 32 KB A-panel: [c][p]
    __shared__ float redmax[16 * 8];
    __shared__ float redsum[16 * 8];

    const int n  = blockIdx.y;
    const int p0 = blockIdx.x * 16;
    const int cl = cls[n];
    const int t    = threadIdx.x;
    const int lane = t & 31;
    const int wave = t >> 5;
    const int ml   = lane & 15;         // N / M-row index inside fragment
    const int hi   = lane >> 4;         // half-wave selector

    // stage A-panel asynchronously: As[c*16 + p] = z[n, c, p0+p]
    for (int idx = t; idx < CC * 16; idx += 256) {
        int c = idx >> 4, p = idx & 15;
        async_g2l_b32(&z[(((size_t)n * CC + c) * HW) + p0 + p], &As[idx]);
    }
    async_wait0();
    __syncthreads();

    float z2v[8];
#pragma unroll
    for (int r = 0; r < 8; ++r) z2v[r] = z2g[n * HW + p0 + r + 8 * hi];

    const float* clbase = memT_cm + (size_t)cl * CM;

    v8f   acc[16];
    float m2t[16];
#pragma unroll
    for (int i = 0; i < 16; ++i) {
        const int tile = wave + 8 * i;
        if (tile < 125) {
            const int m0 = tile * 16;
            const float* Bp = clbase + m0 + ml;        // + c*MM per k row
            __builtin_prefetch(Bp, 0, 1);
            v8f a = {0.f, 0.f, 0.f, 0.f, 0.f, 0.f, 0.f, 0.f};
            for (int c = 0; c < CC; c += 4) {
                v2f af, bf;
                af.x = As[(c + 2 * hi) * 16 + ml];
                af.y = As[(c + 1 + 2 * hi) * 16 + ml];
                bf.x = Bp[(size_t)(c + 2 * hi) * MM];
                bf.y = Bp[(size_t)(c + 1 + 2 * hi) * MM];
                a = __builtin_amdgcn_wmma_f32_16x16x4_f32(
                        false, af, false, bf, (short)0, a, false, false);
            }
            acc[i] = a;
            m2t[i] = m2g[cl * MM + m0 + ml];
        }
    }

    // ---- row max ----
    float mx[8];
#pragma unroll
    for (int r = 0; r < 8; ++r) mx[r] = -3.402823466e38f;
#pragma unroll
    for (int i = 0; i < 16; ++i) {
        if ((wave + 8 * i) < 125) {
#pragma unroll
            for (int r = 0; r < 8; ++r) {
                float s = 2.f * acc[i][r] - z2v[r] - m2t[i];
                mx[r] = fmaxf(mx[r], s);
            }
        }
    }
#pragma unroll
    for (int r = 0; r < 8; ++r) {
#pragma unroll
        for (int off = 1; off < 16; off <<= 1)
            mx[r] = fmaxf(mx[r], __shfl_xor(mx[r], off, 32));
    }
    if (ml == 0) {   // lane 0 (rows 0..7) and lane 16 (rows 8..15)
#pragma unroll
        for (int r = 0; r < 8; ++r) redmax[(r + 8 * hi) * 8 + wave] = mx[r];
    }
    __syncthreads();
    float rowmax[8];
#pragma unroll
    for (int r = 0; r < 8; ++r) {
        float v = redmax[(r + 8 * hi) * 8];
#pragma unroll
        for (int w = 1; w < 8; ++w) v = fmaxf(v, redmax[(r + 8 * hi) * 8 + w]);
        rowmax[r] = v;
    }

    // ---- row sum of exp ----
    float sm[8];
#pragma unroll
    for (int r = 0; r < 8; ++r) sm[r] = 0.f;
#pragma unroll
    for (int i = 0; i < 16; ++i) {
        if ((wave + 8 * i) < 125) {
#pragma unroll
            for (int r = 0; r < 8; ++r) {
                float s = 2.f * acc[i][r] - z2v[r] - m2t[i];
                sm[r] += __expf(s - rowmax[r]);
            }
        }
    }
#pragma unroll
    for (int r = 0; r < 8; ++r) {
#pragma unroll
        for (int off = 1; off < 16; off <<= 1)
            sm[r] += __shfl_xor(sm[r], off, 32);
    }
    if (ml == 0) {
#pragma unroll
        for (int r = 0; r < 8; ++r) redsum[(r + 8 * hi) * 8 + wave] = sm[r];
    }
    __syncthreads();
    float inv[8], lgs[8];
#pragma unroll
    for (int r = 0; r < 8; ++r) {
        float v = 0.f;
#pragma unroll
        for (int w = 0; w < 8; ++w) v += redsum[(r + 8 * hi) * 8 + w];
        inv[r] = 1.f / v;
        lgs[r] = __logf(v);
    }

    // ---- write w_hat, w, log_w ----
#pragma unroll
    for (int i = 0; i < 16; ++i) {
        const int tile = wave + 8 * i;
        if (tile < 125) {
            const int m0 = tile * 16;
#pragma unroll
            for (int r = 0; r < 8; ++r) {
                float s = 2.f * acc[i][r] - z2v[r] - m2t[i];
                float e = __expf(s - rowmax[r]);
                float wv = e * inv[r];
                float lw = s - rowmax[r] - lgs[r];
                size_t idx = ((size_t)(n * HW + p0 + r + 8 * hi)) * MM + m0 + ml;
                out_what[idx] = wv;
                out_w[idx]    = wv;
                out_logw[idx] = lw;
            }
        }
    }
}

// ---------------------------------------------------------------------------
// Kernel 6: GEMM2  z_hat[n,c,p] = sum_m memcls[c,m] * w[n,p,m]
// Block = (sample n, 16-col p-tile).  w strip staged transposed into LDS in
// 4 chunks of 500 m-values (padded stride 17) with DOUBLE BUFFERING via
// async global->LDS: chunk k+1 streams in while chunk k feeds the WMMAs.
// A is read coalesced from the [m][c] copy.
// ---------------------------------------------------------------------------
__global__ void __launch_bounds__(256)
k_gemm2(const int* __restrict__ cls, const float* __restrict__ memT_mc,
        const float* __restrict__ w_in, float* __restrict__ z_hat) {
    __shared__ float Wt[2][500 * 17];   // 2 x 34 KB, [m_local][p] padded

    const int n  = blockIdx.y;
    const int p0 = blockIdx.x * 16;
    const int cl = cls[n];
    const int t    = threadIdx.x;
    const int lane = t & 31;
    const int wave = t >> 5;
    const int ml   = lane & 15;
    const int hi   = lane >> 4;

    const float* mcbase = memT_mc + (size_t)cl * CM;
    const float* wbase  = w_in + (size_t)(n * HW + p0) * MM;

    v8f acc[4];
#pragma unroll
    for (int j = 0; j < 4; ++j) acc[j] = (v8f){0.f,0.f,0.f,0.f,0.f,0.f,0.f,0.f};

    // async-stage chunk 0
    for (int idx = t; idx < 500 * 16; idx += 256) {
        int p = idx / 500, m = idx % 500;
        async_g2l_b32(&wbase[(size_t)p * MM + m], &Wt[0][m * 17 + p]);
    }
    async_wait0();
    __syncthreads();

    for (int chunk = 0; chunk < 4; ++chunk) {
        const int buf = chunk & 1;
        // prefetch next chunk into the other buffer while we compute
        if (chunk < 3) {
            const int nb = chunk + 1;
            for (int idx = t; idx < 500 * 16; idx += 256) {
                int p = idx / 500, m = idx % 500;
                async_g2l_b32(&wbase[(size_t)p * MM + nb * 500 + m],
                              &Wt[nb & 1][m * 17 + p]);
            }
        }
        const int mbase = chunk * 500;
#pragma unroll
        for (int j = 0; j < 4; ++j) {
            const int c0 = (wave + 8 * j) * 16;
            const float* Ap = mcbase + c0 + ml;        // + m*CC per k row
            __builtin_prefetch(Ap + (size_t)mbase * CC, 0, 1);
            v8f a = acc[j];
            for (int k = 0; k < 500; k += 4) {
                const int mg = mbase + k;
                v2f af, bf;
                af.x = Ap[(size_t)(mg + 2 * hi) * CC];
                af.y = Ap[(size_t)(mg + 1 + 2 * hi) * CC];
                bf.x = Wt[buf][(k + 2 * hi) * 17 + ml];
                bf.y = Wt[buf][(k + 1 + 2 * hi) * 17 + ml];
                a = __builtin_amdgcn_wmma_f32_16x16x4_f32(
                        false, af, false, bf, (short)0, a, false, false);
            }
            acc[j] = a;
        }
        __syncthreads();                 // all waves done reading Wt[buf]
        if (chunk < 3) {
            async_wait0();               // next buffer fully landed in LDS
            __syncthreads();
        }
    }

    // write z_hat: D element (c_local = r + 8*hi, p_local = ml)
#pragma unroll
    for (int j = 0; j < 4; ++j) {
        const int c0 = (wave + 8 * j) * 16;
#pragma unroll
        for (int r = 0; r < 8; ++r) {
            size_t idx = ((size_t)n * CC + c0 + r + 8 * hi) * HW + p0 + ml;
            z_hat[idx] = acc[j][r];
        }
    }
}

// ---------------------------------------------------------------------------
// Host launcher
// ---------------------------------------------------------------------------
extern "C" void kernel_launch(void* const* d_in, const int* in_sizes, int n_in,
                              void* d_out, int out_size, void* d_ws, size_t ws_size,
                              hipStream_t stream) {
    const float* z   = (const float*)d_in[0];
    const int*   cls = (const int*)d_in[1];
    const float* mem = (const float*)d_in[2];

    float* out = (float*)d_out;
    float* z_hat = out;                                   // 64*512*256
    float* w_hat = out + (size_t)NN * CC * HW;            // 64*256*2000
    float* w     = w_hat + (size_t)NN * HW * MM;
    float* log_w = w     + (size_t)NN * HW * MM;

    float* wsf      = (float*)d_ws;
    float* memT_cm  = wsf;                                // 10*512*2000
    float* memT_mc  = memT_cm + (size_t)NCLS * CM;        // 10*2000*512
    float* m2       = memT_mc + (size_t)NCLS * CM;        // 10*2000
    float* z2       = m2 + NCLS * MM;                     // 64*256

    // 1. gather per-class planes [c][m]
    k_gather_cm<<<(CM + 255) / 256, 256, 0, stream>>>(mem, memT_cm);
    // 2. transpose to [m][c]
    k_transpose_mc<<<dim3((MM + 31) / 32, CC / 32, NCLS), dim3(32, 8), 0, stream>>>(
        memT_cm, memT_mc);
    // 3. m2 per class
    k_m2<<<(NCLS * MM + 255) / 256, 256, 0, stream>>>(memT_cm, m2);
    // 4. z2 per sample
    k_z2<<<NN, HW, 0, stream>>>(z, z2);
    // 5. fused GEMM1 + softmax -> w_hat, w, log_w
    k_gemm1_softmax<<<dim3(HW / 16, NN), 256, 0, stream>>>(
        z, cls, memT_cm, m2, z2, w_hat, w, log_w);
    // 6. GEMM2 -> z_hat (double-buffered async LDS staging)
    k_gemm2<<<dim3(HW / 16, NN), 256, 0, stream>>>(cls, memT_mc, w_hat, z_hat);
}